// YoloLoss_83940840833481
// MI455X (gfx1250) — compile-verified
//
#include <hip/hip_runtime.h>
#include <hip/hip_bf16.h>

// ---------------------------------------------------------------------------
// YOLO loss, MI455X (gfx1250, wave32).
// Bandwidth-bound: ~176 MB traffic -> ~7.6 us at 23.3 TB/s HBM; inputs fit in
// the 192 MB L2 so graph replays run hotter than HBM. fp32 everywhere (exact);
// V_WMMA_F32_16X16X4_F32 (f32 in, f32 acc) does the exact cross-wave block
// reduction (B = ones => D row m = sum_k A[m,k], pure reassociated f32 adds).
// ---------------------------------------------------------------------------

typedef float v2f __attribute__((ext_vector_type(2)));
typedef float v8f __attribute__((ext_vector_type(8)));

#define THREADS 256
#define INV_S (1.0f / 28.0f)

// Reduce 4 per-thread f32 partials over a 256-thread block.
// On exit, thread 0 (wave 0, lane 0) has written dst[0..3].
__device__ __forceinline__ void block_reduce_store4(float p0, float p1,
                                                    float p2, float p3,
                                                    float* dst) {
  __shared__ float red[32];  // [q*8 + wave], q = 0..3, wave = 0..7
  const int lane = threadIdx.x & 31;
  const int wave = threadIdx.x >> 5;

  // Exact intra-wave tree sum (wave32).
  #pragma unroll
  for (int off = 16; off > 0; off >>= 1) {
    p0 += __shfl_xor(p0, off, 32);
    p1 += __shfl_xor(p1, off, 32);
    p2 += __shfl_xor(p2, off, 32);
    p3 += __shfl_xor(p3, off, 32);
  }
  if (lane == 0) {
    red[0 * 8 + wave] = p0;
    red[1 * 8 + wave] = p1;
    red[2 * 8 + wave] = p2;
    red[3 * 8 + wave] = p3;
  }
  __syncthreads();

  if (wave == 0) {
    // A-matrix (16x4 f32) layout: lanes 0-15 -> M=lane, VGPR0=K0, VGPR1=K1;
    // lanes 16-31 -> M=lane-16, VGPR0=K2, VGPR1=K3.
    // Row m sums the 4 wave-partials of quantity m; two chained WMMAs cover
    // waves 0-3 and 4-7. B = all ones (storage-layout independent).
    const int m = lane & 15;
    const int hi = lane >> 4;
    const bool rowok = (m < 4);
    const int base = (rowok ? m : 0) * 8 + hi * 2;
    v2f aLo = {0.0f, 0.0f};
    v2f aHi = {0.0f, 0.0f};
    const float keep = rowok ? 1.0f : 0.0f;
    aLo[0] = keep * red[base + 0];
    aLo[1] = keep * red[base + 1];
    aHi[0] = keep * red[base + 4];
    aHi[1] = keep * red[base + 5];
    v2f ones;
    ones[0] = 1.0f;
    ones[1] = 1.0f;
    v8f c = {};
    c = __builtin_amdgcn_wmma_f32_16x16x4_f32(false, aLo, false, ones,
                                              (short)0, c, false, false);
    c = __builtin_amdgcn_wmma_f32_16x16x4_f32(false, aHi, false, ones,
                                              (short)0, c, false, false);
    // D layout: VGPR j, lane 0 -> D[M=j][N=0]; rows 0..3 hold the 4 sums.
    if (lane == 0) {
      dst[0] = c[0];
      dst[1] = c[1];
      dst[2] = c[2];
      dst[3] = c[3];
    }
  }
}

__device__ __forceinline__ float safe_sqrtf(float x) {
  return (x > 0.0f) ? __builtin_sqrtf(x) : 0.0f;
}

__global__ void __launch_bounds__(THREADS) yolo_main(
    const float* __restrict__ pred,   // [n,28,28,30]
    const float* __restrict__ tbox,   // [n,28,28,4]
    const float* __restrict__ tcls,   // [n,28,28,20]
    const int* __restrict__ omap,     // [n,28,28] (bool -> int32)
    float* __restrict__ ws,           // [gridDim.x][4] block partials
    int ncells) {
  const int tid = blockIdx.x * THREADS + threadIdx.x;
  const bool live = tid < ncells;
  const size_t t = live ? (size_t)tid : 0;  // clamp, zero contribution later

  // ---- loads (120 B/lane pred is 8B aligned -> b64; box/cls are 16B aligned)
  float pv[30];
  {
    const v2f* p2 = (const v2f*)(pred + t * 30);
    #pragma unroll
    for (int i = 0; i < 15; ++i) {
      v2f v = p2[i];
      pv[2 * i + 0] = v[0];
      pv[2 * i + 1] = v[1];
    }
  }
  const float4 tb = ((const float4*)tbox)[t];
  float tc[20];
  {
    const float4* c4 = (const float4*)tcls + t * 5;
    #pragma unroll
    for (int i = 0; i < 5; ++i) {
      float4 v = c4[i];
      tc[4 * i + 0] = v.x;
      tc[4 * i + 1] = v.y;
      tc[4 * i + 2] = v.z;
      tc[4 * i + 3] = v.w;
    }
  }
  const float m = (omap[t] != 0) ? 1.0f : 0.0f;

  // ---- class loss partial: m * sum_c (pc - tc)^2
  float cls = 0.0f;
  #pragma unroll
  for (int cidx = 0; cidx < 20; ++cidx) {
    const float d = pv[10 + cidx] - tc[cidx];
    cls = __builtin_fmaf(d, d, cls);
  }
  cls *= m;

  // ---- no-object partial: (1-m) * (conf0^2 + conf1^2)
  const float nbj = (1.0f - m) * (pv[4] * pv[4] + pv[9] * pv[9]);

  // ---- target box xyxy
  const float tx = tb.x * INV_S, ty = tb.y * INV_S;
  const float thw = 0.5f * tb.z, thh = 0.5f * tb.w;
  const float tx0 = tx - thw, ty0 = ty - thh;
  const float tx1 = tx + thw, ty1 = ty + thh;
  const float ta = (tx1 - tx0) * (ty1 - ty0);

  // ---- per-pred-box IoU (B = 2)
  float bx0[2], by0[2], bx1[2], by1[2], iou[2];
  #pragma unroll
  for (int b = 0; b < 2; ++b) {
    const int o = 5 * b;
    const float px = pv[o + 0] * INV_S, py = pv[o + 1] * INV_S;
    const float phw = 0.5f * pv[o + 2], phh = 0.5f * pv[o + 3];
    const float x0 = px - phw, y0 = py - phh;
    const float x1 = px + phw, y1 = py + phh;
    bx0[b] = x0; by0[b] = y0; bx1[b] = x1; by1[b] = y1;
    const float iw = fmaxf(fminf(x1, tx1) - fmaxf(x0, tx0), 0.0f);
    const float ih = fmaxf(fminf(y1, ty1) - fmaxf(y0, ty0), 0.0f);
    const float inter = iw * ih;
    const float a1 = (x1 - x0) * (y1 - y0);
    iou[b] = inter / (a1 + ta - inter);
  }

  // ---- best box (argmax, first on tie; NaN compares false -> picks box 0,
  //      then valid test rejects it, matching jnp semantics)
  const bool pick1 = iou[1] > iou[0];
  float biou = pick1 ? iou[1] : iou[0];
  float sx0 = pick1 ? bx0[1] : bx0[0];
  float sy0 = pick1 ? by0[1] : by0[0];
  float sx1 = pick1 ? bx1[1] : bx1[0];
  float sy1 = pick1 ? by1[1] : by1[0];
  float bconf = pick1 ? pv[9] : pv[4];
  const bool valid = biou > 0.0f;
  biou = valid ? biou : 0.0f;
  sx0 = valid ? sx0 : 0.0f;
  sy0 = valid ? sy0 : 0.0f;
  sx1 = valid ? sx1 : 0.0f;
  sy1 = valid ? sy1 : 0.0f;
  bconf = valid ? bconf : 0.0f;

  // ---- regression + conf-obj partials
  const float dx = sx0 - tx0, dy = sy0 - ty0;
  const float lxy = dx * dx + dy * dy;
  const float dwx = safe_sqrtf(sx1) - safe_sqrtf(tx1);
  const float dwy = safe_sqrtf(sy1) - safe_sqrtf(ty1);
  const float lwh = dwx * dwx + dwy * dwy;
  float reg = m * (lxy + lwh);
  const float dc = bconf - biou;
  float cobj = m * dc * dc;

  const float alive = live ? 1.0f : 0.0f;
  block_reduce_store4(cls * alive, nbj * alive, reg * alive, cobj * alive,
                      ws + (size_t)blockIdx.x * 4);
}

__global__ void __launch_bounds__(THREADS) yolo_finalize(
    const float* __restrict__ ws, float* __restrict__ out, int nparts) {
  float p0 = 0.0f, p1 = 0.0f, p2 = 0.0f, p3 = 0.0f;
  for (int j = threadIdx.x; j < nparts; j += THREADS) {
    p0 += ws[j * 4 + 0];
    p1 += ws[j * 4 + 1];
    p2 += ws[j * 4 + 2];
    p3 += ws[j * 4 + 3];
  }
  __shared__ float res[4];
  block_reduce_store4(p0, p1, p2, p3, res);
  // thread 0 wrote res itself; no extra sync needed for its own read
  if (threadIdx.x == 0) {
    const float invN = 1.0f / 1024.0f;
    const float cls_l = 2.0f * res[0] * invN;
    const float nbj_l = 0.5f * res[1] * invN;
    const float reg_l = 5.0f * res[2] * invN;
    const float cobj_l = res[3] * invN;
    out[0] = cls_l + nbj_l + reg_l + cobj_l;  // total
    out[1] = reg_l;
    out[2] = cobj_l;
    out[3] = nbj_l;
    out[4] = cls_l;
  }
}

extern "C" void kernel_launch(void* const* d_in, const int* in_sizes, int n_in,
                              void* d_out, int out_size, void* d_ws,
                              size_t ws_size, hipStream_t stream) {
  const float* pred = (const float*)d_in[0];
  const float* tbox = (const float*)d_in[1];
  const float* tcls = (const float*)d_in[2];
  const int* omap = (const int*)d_in[3];
  float* out = (float*)d_out;
  float* ws = (float*)d_ws;

  const int ncells = in_sizes[3];               // 1024*28*28 = 802816
  const int nblk = (ncells + THREADS - 1) / THREADS;  // 3136

  yolo_main<<<nblk, THREADS, 0, stream>>>(pred, tbox, tcls, omap, ws, ncells);
  yolo_finalize<<<1, THREADS, 0, stream>>>(ws, out, nblk);
}